// cfrm_best_5205500362863
// MI455X (gfx1250) — compile-verified
//
#include <hip/hip_runtime.h>
#include <math.h>

// ---------------------------------------------------------------------------
// Problem constants (from the reference)
// ---------------------------------------------------------------------------
#define G_DIM 8000
#define R_DIM 20000
#define M1 1200
#define M2 1200
#define MG 2400
#define MR 6000

#define TJ1 (MG / 16)            // 150
#define TJ2 (MR / 16)            // 375
#define TJ3 (MR / 16)            // 375
#define NT1 ((M1 / 16) * TJ1)    // 11250 tiles
#define NT2 ((M2 / 16) * TJ2)    // 28125 tiles
#define NT3 ((MG / 16) * TJ3)    // 56250 tiles

#define WAVES_PER_BLOCK 8
#define NB1 ((NT1 + WAVES_PER_BLOCK - 1) / WAVES_PER_BLOCK)  // 1407
#define NB2 ((NT2 + WAVES_PER_BLOCK - 1) / WAVES_PER_BLOCK)  // 3516
#define NB3 ((NT3 + WAVES_PER_BLOCK - 1) / WAVES_PER_BLOCK)  // 7032

typedef __attribute__((ext_vector_type(2))) float v2f;
typedef __attribute__((ext_vector_type(8))) float v8f;

// ---------------------------------------------------------------------------
// Prep: gather masked softmax rows (optionally times Ai), pad NF=3 -> 4
// ---------------------------------------------------------------------------
__global__ void prep_left(const float* __restrict__ C, const float* __restrict__ Ai,
                          const float* __restrict__ bvec, const int* __restrict__ mask,
                          int M, float* __restrict__ Q, float* __restrict__ bout) {
    int i = blockIdx.x * blockDim.x + threadIdx.x;
    if (i >= M) return;
    int r = mask[i];
    float x0 = C[r * 3 + 0], x1 = C[r * 3 + 1], x2 = C[r * 3 + 2];
    float mx = fmaxf(x0, fmaxf(x1, x2));
    float e0 = expf(x0 - mx), e1 = expf(x1 - mx), e2 = expf(x2 - mx);
    float inv = 1.0f / (e0 + e1 + e2);
    float p0 = e0 * inv, p1 = e1 * inv, p2 = e2 * inv;
    // Q = P @ Ai  (Ai row-major 3x3)
    Q[i * 4 + 0] = p0 * Ai[0] + p1 * Ai[3] + p2 * Ai[6];
    Q[i * 4 + 1] = p0 * Ai[1] + p1 * Ai[4] + p2 * Ai[7];
    Q[i * 4 + 2] = p0 * Ai[2] + p1 * Ai[5] + p2 * Ai[8];
    Q[i * 4 + 3] = 0.0f;
    bout[i] = bvec[r];
}

__global__ void prep_right(const float* __restrict__ C, const float* __restrict__ bvec,
                           const int* __restrict__ mask, int M,
                           float* __restrict__ P, float* __restrict__ bout) {
    int i = blockIdx.x * blockDim.x + threadIdx.x;
    if (i >= M) return;
    int r = mask[i];
    float x0 = C[r * 3 + 0], x1 = C[r * 3 + 1], x2 = C[r * 3 + 2];
    float mx = fmaxf(x0, fmaxf(x1, x2));
    float e0 = expf(x0 - mx), e1 = expf(x1 - mx), e2 = expf(x2 - mx);
    float inv = 1.0f / (e0 + e1 + e2);
    P[i * 4 + 0] = e0 * inv;
    P[i * 4 + 1] = e1 * inv;
    P[i * 4 + 2] = e2 * inv;
    P[i * 4 + 3] = 0.0f;
    bout[i] = bvec[r];
}

// ---------------------------------------------------------------------------
// Main tile kernel: one wave per 16x16 output tile.
// pred tile = Ltile[16x4(pad)] x Rtile^T[4x16]  via V_WMMA_F32_16X16X4_F32,
// then fuse the gathered element, biases, square (or product) and reduce.
// SQUARED=true:  acc += (X - pred - bcol - brow)^2     (loss1/loss2)
// SQUARED=false: acc += X * pred                        (loss3, unnormalized)
// ---------------------------------------------------------------------------
template <bool SQUARED>
__global__ void tile_kernel(const float* __restrict__ X, int ld,
                            const int* __restrict__ rowmask, const int* __restrict__ colmask,
                            const float* __restrict__ L, const float* __restrict__ Rv,
                            const float* __restrict__ brow, const float* __restrict__ bcol,
                            float* __restrict__ partial, int ntiles, int tilesJ) {
    const int wave = threadIdx.x >> 5;
    const int lane = threadIdx.x & 31;
    const int t = blockIdx.x * WAVES_PER_BLOCK + wave;
    float acc = 0.0f;
    if (t < ntiles) {   // wave-uniform branch: EXEC stays all-ones for WMMA
        const int ti = t / tilesJ;
        const int tj = t % tilesJ;
        const int half = lane >> 4;      // lane 16..31 -> upper K / M halves
        const int l15 = lane & 15;
        const int kb = half * 2;         // K base for this lane's 2 VGPRs

        // A fragment: 16x4 f32 (Q rows, padded).  lane: M=l15, K=kb..kb+1
        v2f a = *(const v2f*)(L + (size_t)(ti * 16 + l15) * 4 + kb);
        // B fragment: 4x16 f32 (P rows as columns). lane: N=l15, K=kb..kb+1
        v2f b = *(const v2f*)(Rv + (size_t)(tj * 16 + l15) * 4 + kb);

        v8f c = {};
        c = __builtin_amdgcn_wmma_f32_16x16x4_f32(
                /*neg_a=*/false, a, /*neg_b=*/false, b,
                /*c_mod=*/(short)0, c, /*reuse_a=*/false, /*reuse_b=*/false);

        // D layout: VGPR v, lane -> (M = v + 8*half, N = l15)
        const int j = tj * 16 + l15;          // column fixed per lane
        const int cj = colmask[j];
        const float bc = SQUARED ? bcol[j] : 0.0f;
#pragma unroll
        for (int v = 0; v < 8; ++v) {
            const int i = ti * 16 + v + 8 * half;
            const int ri = rowmask[i];
            const float x = X[(size_t)ri * (size_t)ld + (size_t)cj];
            if (SQUARED) {
                const float d = x - c[v] - bc - brow[i];
                acc += d * d;
            } else {
                acc += x * c[v];
            }
        }
    }
    // wave32 reduction, then fixed-order combine across waves (deterministic)
#pragma unroll
    for (int off = 16; off > 0; off >>= 1) acc += __shfl_xor(acc, off, 32);
    __shared__ float wsum[WAVES_PER_BLOCK];
    if (lane == 0) wsum[wave] = acc;
    __syncthreads();
    if (threadIdx.x == 0) {
        float s = 0.0f;
        for (int w = 0; w < WAVES_PER_BLOCK; ++w) s += wsum[w];
        partial[blockIdx.x] = s;
    }
}

// ---------------------------------------------------------------------------
// Finalize: single block, deterministic tree reductions + alpha combine
// ---------------------------------------------------------------------------
__global__ void finalize_kernel(const float* __restrict__ p1, int n1,
                                const float* __restrict__ p2, int n2,
                                const float* __restrict__ p3, int n3,
                                const float* __restrict__ Pg, const float* __restrict__ Pr,
                                float* __restrict__ out) {
    __shared__ float red[256];
    const int tid = threadIdx.x;
    auto reduce = [&](float v) -> float {
        red[tid] = v;
        __syncthreads();
        for (int off = 128; off > 0; off >>= 1) {
            if (tid < off) red[tid] += red[tid + off];
            __syncthreads();
        }
        float r = red[0];
        __syncthreads();
        return r;
    };
    float s;
    s = 0.0f; for (int i = tid; i < n1; i += 256) s += p1[i];
    const float sum1 = reduce(s);
    s = 0.0f; for (int i = tid; i < n2; i += 256) s += p2[i];
    const float sum2 = reduce(s);
    s = 0.0f; for (int i = tid; i < n3; i += 256) s += p3[i];
    const float sum3 = reduce(s);
    s = 0.0f;
    for (int i = tid; i < MG; i += 256) {
        float a = Pg[i * 4], b = Pg[i * 4 + 1], c = Pg[i * 4 + 2];
        s += a * a + b * b + c * c;
    }
    const float sgg = reduce(s);
    s = 0.0f;
    for (int i = tid; i < MR; i += 256) {
        float a = Pr[i * 4], b = Pr[i * 4 + 1], c = Pr[i * 4 + 2];
        s += a * a + b * b + c * c;
    }
    const float srr = reduce(s);
    if (tid == 0) {
        const float l1 = 1000.0f * sum1 / (float)(M1 * MG);
        const float l2 = 1000.0f * sum2 / (float)(M2 * MR);
        const float l3 = -100.0f * sum3 / (sqrtf(sgg) * sqrtf(srr));
        out[0] = l1 + l2 + l3;   // ALPHA[3] == 0 -> entropy term contributes 0
        out[1] = l1;
        out[2] = l2;
        out[3] = l3;
        out[4] = 0.0f;
    }
}

// ---------------------------------------------------------------------------
// Host launch
// ---------------------------------------------------------------------------
extern "C" void kernel_launch(void* const* d_in, const int* in_sizes, int n_in,
                              void* d_out, int out_size, void* d_ws, size_t ws_size,
                              hipStream_t stream) {
    (void)in_sizes; (void)n_in; (void)out_size; (void)ws_size;
    const float* G  = (const float*)d_in[0];
    const float* R  = (const float*)d_in[1];
    const float* A  = (const float*)d_in[2];
    const float* C1 = (const float*)d_in[3];
    const float* C2 = (const float*)d_in[4];
    const float* Cg = (const float*)d_in[5];
    const float* Cr = (const float*)d_in[6];
    const float* Ai = (const float*)d_in[7];
    const float* bg = (const float*)d_in[8];
    const float* br = (const float*)d_in[9];
    const float* b1 = (const float*)d_in[10];
    const float* b2 = (const float*)d_in[11];
    const int* m1 = (const int*)d_in[12];
    const int* m2 = (const int*)d_in[13];
    const int* mg = (const int*)d_in[14];
    const int* mr = (const int*)d_in[15];

    float* ws = (float*)d_ws;
    float* Q1  = ws; ws += M1 * 4;
    float* Q2  = ws; ws += M2 * 4;
    float* Pg  = ws; ws += MG * 4;
    float* Pr  = ws; ws += MR * 4;
    float* b1g = ws; ws += M1;
    float* b2g = ws; ws += M2;
    float* bgg = ws; ws += MG;
    float* brg = ws; ws += MR;
    float* part1 = ws; ws += NB1;
    float* part2 = ws; ws += NB2;
    float* part3 = ws; ws += NB3;

    prep_left <<<(M1 + 255) / 256, 256, 0, stream>>>(C1, Ai, b1, m1, M1, Q1, b1g);
    prep_left <<<(M2 + 255) / 256, 256, 0, stream>>>(C2, Ai, b2, m2, M2, Q2, b2g);
    prep_right<<<(MG + 255) / 256, 256, 0, stream>>>(Cg, bg, mg, MG, Pg, bgg);
    prep_right<<<(MR + 255) / 256, 256, 0, stream>>>(Cr, br, mr, MR, Pr, brg);

    tile_kernel<true ><<<NB1, 256, 0, stream>>>(G, G_DIM, m1, mg, Q1, Pg, b1g, bgg, part1, NT1, TJ1);
    tile_kernel<true ><<<NB2, 256, 0, stream>>>(R, R_DIM, m2, mr, Q2, Pr, b2g, brg, part2, NT2, TJ2);
    tile_kernel<false><<<NB3, 256, 0, stream>>>(A, R_DIM, mg, mr, Pg, Pr, nullptr, nullptr, part3, NT3, TJ3);

    finalize_kernel<<<1, 256, 0, stream>>>(part1, NB1, part2, NB2, part3, NB3, Pg, Pr, (float*)d_out);
}